// EdgeEmbed_21088289424017
// MI455X (gfx1250) — compile-verified
//
#include <hip/hip_runtime.h>

typedef float v2f __attribute__((ext_vector_type(2)));
typedef float v4f __attribute__((ext_vector_type(4)));
typedef float v8f __attribute__((ext_vector_type(8)));

#define DIM     512
#define DE      64
#define NTOK    768
#define BATCH   2
#define MROWS   (BATCH * NTOK)   // 1536
#define NCOMB   (2 * DE)         // 128 (R2 cols 0..63, C2 cols 64..127)
#define LN_EPS  1e-5f

// ---------------------------------------------------------------------------
// Kernel 1: fold W_edges into the row/col projections.
//   Wcomb[f][d]    = sum_e W_edges[f][e] * W_rows[e][d]   (f in 0..63)
//   Wcomb[64+f][d] = sum_e W_edges[f][e] * W_cols[e][d]
// ---------------------------------------------------------------------------
__global__ void fuse_weights_kernel(const float* __restrict__ W_rows,
                                    const float* __restrict__ W_cols,
                                    const float* __restrict__ W_edges,
                                    float* __restrict__ Wcomb) {
    int idx = blockIdx.x * blockDim.x + threadIdx.x;   // 0 .. 128*512-1
    int f2  = idx >> 9;                                // / 512
    int d   = idx & (DIM - 1);
    const float* W = (f2 < DE) ? W_rows : W_cols;
    int f = f2 & (DE - 1);
    float acc = 0.0f;
    #pragma unroll 8
    for (int e = 0; e < DE; ++e)
        acc += W_edges[f * DE + e] * W[e * DIM + d];
    Wcomb[idx] = acc;
}

// ---------------------------------------------------------------------------
// Kernel 2: RC[1536,128] = X[1536,512] @ Wcomb[128,512]^T
// via V_WMMA_F32_16X16X4_F32. One wave per 16x16 tile, K-loop of 128 WMMAs.
// A (16x4 MxK) f32 layout: lanes 0-15 M=lane K={0,1}; lanes 16-31 K={2,3}.
// B (4x16 KxN) mirrors with N across lanes in each half.
// C/D: VGPR r -> M = r + 8*(lane>=16), N = lane&15.
// ---------------------------------------------------------------------------
__global__ void gemm_rc_wmma_kernel(const float* __restrict__ X,    // [1536,512]
                                    const float* __restrict__ Wc,   // [128,512]
                                    float* __restrict__ RC) {       // [1536,128]
    int wave  = (blockIdx.x * blockDim.x + threadIdx.x) >> 5;
    int lane  = threadIdx.x & 31;
    int mTile = wave >> 3;          // 96 M-tiles
    int nTile = wave & 7;           // 8  N-tiles
    int m0 = mTile * 16;
    int n0 = nTile * 16;

    int half = lane >> 4;
    int lm   = lane & 15;

    const float* arow = X  + (size_t)(m0 + lm) * DIM + 2 * half;
    const float* brow = Wc + (size_t)(n0 + lm) * DIM + 2 * half;

    v8f c = {};
    #pragma unroll 4
    for (int k = 0; k < DIM; k += 4) {
        v2f a = *(const v2f*)(arow + k);
        v2f b = *(const v2f*)(brow + k);
        c = __builtin_amdgcn_wmma_f32_16x16x4_f32(
                false, a, false, b, (short)0, c, false, false);
    }

    #pragma unroll
    for (int r = 0; r < 8; ++r) {
        int row = m0 + r + 8 * half;
        int col = n0 + lm;
        RC[(size_t)row * NCOMB + col] = c[r];
    }
}

// ---------------------------------------------------------------------------
// Kernel 2b: per-row first/second moments of R2 and C2.
//   RowStat[row] = { S_r, Q_r, S_c, Q_c }   (row = b*768+i, 1536 rows)
// ---------------------------------------------------------------------------
__global__ void row_moments_kernel(const float* __restrict__ RC,   // [1536,128]
                                   float* __restrict__ RowStat) {  // [1536,4]
    int row = blockIdx.x * blockDim.x + threadIdx.x;
    if (row >= MROWS) return;
    const float* p = RC + (size_t)row * NCOMB;
    float sr = 0.f, qr = 0.f, sc = 0.f, qc = 0.f;
    #pragma unroll
    for (int f = 0; f < DE; f += 4) {
        v4f r = *(const v4f*)(p + f);
        v4f c = *(const v4f*)(p + DE + f);
        sr += r.x + r.y + r.z + r.w;
        qr += r.x * r.x + r.y * r.y + r.z * r.z + r.w * r.w;
        sc += c.x + c.y + c.z + c.w;
        qc += c.x * c.x + c.y * c.y + c.z * c.z + c.w * c.w;
    }
    v4f st = { sr, qr, sc, qc };
    *(v4f*)(RowStat + 4 * row) = st;
}

// ---------------------------------------------------------------------------
// Kernel 2c: per-pair LN statistics via WMMA cross-dot.
//   Dot[b,i,j] = sum_f R2[b,i,f] * C2[b,j,f]   (GEMM M=N=768, K=64 per batch)
//   mu  = (S_r + S_c)/64
//   E2  = (Q_r + 2*Dot + Q_c)/64
//   a   = rsqrt(E2 - mu^2 + eps),  bcoef = -mu*a
//   Stats[pair] = {a, bcoef}
// One wave per 16x16 (i,j) tile; 16 WMMAs over K=64.
// ---------------------------------------------------------------------------
__global__ void pair_stats_wmma_kernel(const float* __restrict__ RC,      // [1536,128]
                                       const float* __restrict__ RowStat, // [1536,4]
                                       float* __restrict__ Stats) {       // [2*768*768,2]
    int wave = (blockIdx.x * blockDim.x + threadIdx.x) >> 5;
    int lane = threadIdx.x & 31;
    const int TPB = (NTOK / 16) * (NTOK / 16);    // 2304 tiles per batch
    int b    = wave / TPB;
    int rem  = wave - b * TPB;
    int iT   = rem / (NTOK / 16);
    int jT   = rem - iT * (NTOK / 16);
    int i0 = iT * 16, j0 = jT * 16;

    int half = lane >> 4;
    int lm   = lane & 15;

    // A: R2 tile rows (i), B: C2 tile rows (j) as KxN with K=f.
    const float* arow = RC + (size_t)(b * NTOK + i0 + lm) * NCOMB + 2 * half;
    const float* brow = RC + (size_t)(b * NTOK + j0 + lm) * NCOMB + DE + 2 * half;

    v8f d = {};
    #pragma unroll
    for (int k = 0; k < DE; k += 4) {
        v2f a  = *(const v2f*)(arow + k);
        v2f bb = *(const v2f*)(brow + k);
        d = __builtin_amdgcn_wmma_f32_16x16x4_f32(
                false, a, false, bb, (short)0, d, false, false);
    }

    int j = j0 + lm;                                  // N = lane&15
    v4f cst = *(const v4f*)(RowStat + 4 * (b * NTOK + j));  // {S_c,Q_c in .z,.w}
    #pragma unroll
    for (int r = 0; r < 8; ++r) {
        int i = i0 + r + 8 * half;
        v4f rst = *(const v4f*)(RowStat + 4 * (b * NTOK + i));
        float mu  = (rst.x + cst.z) * (1.0f / DE);
        float e2  = (rst.y + 2.0f * d[r] + cst.w) * (1.0f / DE);
        float var = e2 - mu * mu;
        float a   = rsqrtf(var + LN_EPS);
        v2f st = { a, -mu * a };
        size_t pair = ((size_t)(b * NTOK + i)) * NTOK + j;
        *(v2f*)(Stats + 2 * pair) = st;
    }
}

// ---------------------------------------------------------------------------
// Kernel 3: streaming LN apply + store (the 302 MB bandwidth kernel).
//   out[pair, f] = (r_f + c_f) * (a*gamma_f) + (bcoef*gamma_f + beta_f)
// 16 lanes per pair, float4 per lane; no cross-lane ops; NT b128 stores.
// ---------------------------------------------------------------------------
__global__ void edge_apply_kernel(const float* __restrict__ RC,     // [1536,128]
                                  const float* __restrict__ Stats,  // [pairs,2]
                                  const float* __restrict__ gamma,  // [64]
                                  const float* __restrict__ beta,   // [64]
                                  float* __restrict__ out) {
    int t    = blockIdx.x * blockDim.x + threadIdx.x;
    int pair = t >> 4;
    int l    = t & 15;

    int bi = pair / NTOK;              // b*768 + i
    int j  = pair - bi * NTOK;
    int b  = (bi >= NTOK) ? 1 : 0;

    v4f rv = *(const v4f*)(RC + (size_t)bi * NCOMB + 4 * l);
    v4f cv = *(const v4f*)(RC + (size_t)(b * NTOK + j) * NCOMB + DE + 4 * l);
    v2f st = *(const v2f*)(Stats + 2 * (size_t)pair);   // broadcast across group
    v4f g  = *(const v4f*)(gamma + 4 * l);              // L1-resident
    v4f be = *(const v4f*)(beta  + 4 * l);

    v4f ag, bb, o;
    ag.x = st.x * g.x;  ag.y = st.x * g.y;  ag.z = st.x * g.z;  ag.w = st.x * g.w;
    bb.x = __builtin_fmaf(st.y, g.x, be.x);
    bb.y = __builtin_fmaf(st.y, g.y, be.y);
    bb.z = __builtin_fmaf(st.y, g.z, be.z);
    bb.w = __builtin_fmaf(st.y, g.w, be.w);
    o.x = __builtin_fmaf(rv.x + cv.x, ag.x, bb.x);
    o.y = __builtin_fmaf(rv.y + cv.y, ag.y, bb.y);
    o.z = __builtin_fmaf(rv.z + cv.z, ag.z, bb.z);
    o.w = __builtin_fmaf(rv.w + cv.w, ag.w, bb.w);

    v4f* dst = (v4f*)(out + (size_t)pair * DE + 4 * l);
    __builtin_nontemporal_store(o, dst);
}

// ---------------------------------------------------------------------------
extern "C" void kernel_launch(void* const* d_in, const int* in_sizes, int n_in,
                              void* d_out, int out_size, void* d_ws, size_t ws_size,
                              hipStream_t stream) {
    const float* x       = (const float*)d_in[0];  // [2,768,512]
    const float* W_rows  = (const float*)d_in[1];  // [64,512]
    const float* W_cols  = (const float*)d_in[2];  // [64,512]
    const float* W_edges = (const float*)d_in[3];  // [64,64]
    const float* gamma   = (const float*)d_in[4];  // [64]
    const float* beta    = (const float*)d_in[5];  // [64]
    float* out = (float*)d_out;

    // Workspace layout (~10.5 MB total):
    float* Wcomb   = (float*)d_ws;                  // [128,512]   = 256 KB
    float* RC      = Wcomb + NCOMB * DIM;           // [1536,128]  = 768 KB
    float* RowStat = RC + (size_t)MROWS * NCOMB;    // [1536,4]    =  24 KB
    float* Stats   = RowStat + (size_t)MROWS * 4;   // [2*768*768,2] = 9.44 MB

    // 1) fold edge weights into projections
    fuse_weights_kernel<<<(NCOMB * DIM) / 256, 256, 0, stream>>>(
        W_rows, W_cols, W_edges, Wcomb);

    // 2) WMMA GEMM: RC = x @ Wcomb^T  (96x8 tiles, 8 waves/block)
    gemm_rc_wmma_kernel<<<96, 256, 0, stream>>>(x, Wcomb, RC);

    // 2b) per-row moments
    row_moments_kernel<<<(MROWS + 255) / 256, 256, 0, stream>>>(RC, RowStat);

    // 2c) WMMA cross-dot + per-pair LN statistics
    //     2 batches * 48*48 tiles = 4608 waves, 8 waves/block -> 576 blocks
    pair_stats_wmma_kernel<<<576, 256, 0, stream>>>(RC, RowStat, Stats);

    // 3) streaming apply: 2*768*768 pairs * 16 lanes
    const long long total = (long long)BATCH * NTOK * NTOK * 16;
    edge_apply_kernel<<<(int)(total / 256), 256, 0, stream>>>(
        RC, Stats, gamma, beta, out);
}